// SurvivalCopula_13752485282010
// MI455X (gfx1250) — compile-verified
//
#include <hip/hip_runtime.h>

typedef float v2f __attribute__((ext_vector_type(2)));
typedef float v4f __attribute__((ext_vector_type(4)));
typedef float v8f __attribute__((ext_vector_type(8)));

#define GRID 1024
#define NW   8              // waves per block
#define BLK  256            // NW * 32
#define SUPER (NW * 32)     // rows per block iteration

// Fold the linear two-layer nets into effective 64-vector weights + scalar biases.
// wEff layout: [0..63]=w_t, [64..127]=w_c, [128]=b_t, [129]=b_c
__global__ void copula_prep(const float* __restrict__ Wt1, const float* __restrict__ bt1,
                            const float* __restrict__ Wt2, const float* __restrict__ bt2,
                            const float* __restrict__ Wc1, const float* __restrict__ bc1,
                            const float* __restrict__ Wc2, const float* __restrict__ bc2,
                            float* __restrict__ wEff) {
  int f = threadIdx.x;
  if (f < 64) {
    float st = 0.f, sc = 0.f;
#pragma unroll
    for (int h = 0; h < 32; ++h) {
      st += Wt1[f * 32 + h] * Wt2[h];
      sc += Wc1[f * 32 + h] * Wc2[h];
    }
    wEff[f] = st;
    wEff[64 + f] = sc;
  }
  if (f == 0) {
    float bt = bt2[0], bc = bc2[0];
    for (int h = 0; h < 32; ++h) { bt += bt1[h] * Wt2[h]; bc += bc1[h] * Wc2[h]; }
    wEff[128] = bt;
    wEff[129] = bc;
  }
}

struct CopulaParams {
  float shapeT, lscT, lshT;
  float shapeC, lscC, lshC;
  float bT, bC;
};

// One 32-row tile for one wave. K is permuted: hardware slot (step s, half h,
// elem j) carries global k = h*32 + 2s + j for BOTH A and B, making each
// lane's A data 32 consecutive floats (8 x b128 loads per row-half).
template <bool GUARD>
__device__ __forceinline__ float copula_tile(
    int rowBase, const float* __restrict__ x, const float* __restrict__ t,
    const float* __restrict__ cc, int N, const v2f* bfrag,
    int lane, int wave, int n, int half,
    float (*riskLds)[2][32], const CopulaParams& P) {
  const int rA0 = rowBase + n;        // rows 0..15 of tile
  const int rA1 = rowBase + 16 + n;   // rows 16..31
  const float* pa0 = x + (size_t)rA0 * 64 + half * 32;
  const float* pa1 = x + (size_t)rA1 * 64 + half * 32;

  v4f A0[8], A1[8];
  if (GUARD) {
    const v4f z4 = {0.f, 0.f, 0.f, 0.f};
    const bool ok0 = rA0 < N;
    const bool ok1 = rA1 < N;
#pragma unroll
    for (int q = 0; q < 8; ++q) {
      A0[q] = ok0 ? *(const v4f*)(pa0 + q * 4) : z4;
      A1[q] = ok1 ? *(const v4f*)(pa1 + q * 4) : z4;
    }
  } else {
#pragma unroll
    for (int q = 0; q < 8; ++q) {
      A0[q] = *(const v4f*)(pa0 + q * 4);
      A1[q] = *(const v4f*)(pa1 + q * 4);
    }
  }

  v8f c0 = {}, c1 = {};
#pragma unroll
  for (int s = 0; s < 16; ++s) {
    v2f a0 = {A0[s >> 1][(s & 1) * 2], A0[s >> 1][(s & 1) * 2 + 1]};
    v2f a1 = {A1[s >> 1][(s & 1) * 2], A1[s >> 1][(s & 1) * 2 + 1]};
    c0 = __builtin_amdgcn_wmma_f32_16x16x4_f32(false, a0, false, bfrag[s], (short)0, c0, false, false);
    c1 = __builtin_amdgcn_wmma_f32_16x16x4_f32(false, a1, false, bfrag[s], (short)0, c1, false, false);
  }

  __syncthreads();   // protect LDS region against previous iteration's readers
  if (n < 2) {       // lanes 0,1,16,17: scatter C columns 0 (risk_t) and 1 (risk_c)
#pragma unroll
    for (int i = 0; i < 8; ++i) {
      riskLds[wave][n][half * 8 + i]      = c0[i];
      riskLds[wave][n][16 + half * 8 + i] = c1[i];
    }
  }
  __syncthreads();

  const int row = rowBase + lane;
  float out = 0.f;
  if (!GUARD || row < N) {
    float rt = riskLds[wave][0][lane] + P.bT;
    float rc = riskLds[wave][1][lane] + P.bC;
    float tv = t[row], cv = cc[row];
    float lt  = __logf(tv);
    float uT  = rt + P.shapeT * (lt - P.lscT);   // risk + shape*(log t - log scale)
    float uC  = rc + P.shapeC * (lt - P.lscC);
    float lsT = -__expf(uT);                     // log survival
    float lsC = -__expf(uC);
    float ldT = uT - lt + P.lshT + lsT;          // log density
    float ldC = uC - lt + P.lshC + lsC;
    // Clayton copula gradient, theta=2:
    // 1+s = S_E^-2 + S_C^-2 - 1 ; log g_k = -1.5*log(1+s) - 3*ls_k
    float ops = __expf(-2.f * lsT) + __expf(-2.f * lsC) - 1.f;
    float lg  = -1.5f * __logf(ops);
    out = cv * (ldT + (lg - 3.f * lsT)) + (1.f - cv) * (ldC + (lg - 3.f * lsC));
  }
  return out;
}

__global__ __launch_bounds__(BLK) void copula_main(
    const float* __restrict__ x, const float* __restrict__ t, const float* __restrict__ cc,
    const float* __restrict__ wEff,
    const float* __restrict__ pShapeT, const float* __restrict__ pScaleT,
    const float* __restrict__ pShapeC, const float* __restrict__ pScaleC,
    float* __restrict__ partial, int N) {
  __shared__ float riskLds[NW][2][32];
  __shared__ float wsum[NW];

  const int lane = threadIdx.x & 31;
  const int wave = threadIdx.x >> 5;
  const int n    = lane & 15;   // B column / A row-in-tile
  const int half = lane >> 4;

  CopulaParams P;
  P.shapeT = pShapeT[0]; P.shapeC = pShapeC[0];
  P.lscT = __logf(pScaleT[0]); P.lshT = __logf(P.shapeT);
  P.lscC = __logf(pScaleC[0]); P.lshC = __logf(P.shapeC);
  P.bT = wEff[128]; P.bC = wEff[129];

  // B fragment (K-permuted): column 0 = w_t, column 1 = w_c, rest zero.
  v2f bfrag[16];
  {
    const float* wsel = wEff + n * 64 + half * 32;
    const v2f z = {0.f, 0.f};
#pragma unroll
    for (int s = 0; s < 16; ++s)
      bfrag[s] = (n < 2) ? *(const v2f*)(wsel + s * 2) : z;
  }

  const int nFull = N / SUPER;              // fully in-range super-tiles
  const bool hasTail = (N % SUPER) != 0;

  float accL = 0.f;
  for (int st = blockIdx.x; st < nFull; st += GRID)
    accL += copula_tile<false>(st * SUPER + wave * 32, x, t, cc, N, bfrag,
                               lane, wave, n, half, riskLds, P);

  // Tail super-tile (block-uniform condition -> __syncthreads inside is safe)
  if (hasTail && (nFull % GRID) == (int)blockIdx.x)
    accL += copula_tile<true>(nFull * SUPER + wave * 32, x, t, cc, N, bfrag,
                              lane, wave, n, half, riskLds, P);

  // wave32 reduction
#pragma unroll
  for (int off = 16; off > 0; off >>= 1)
    accL += __shfl_xor(accL, off, 32);
  if (lane == 0) wsum[wave] = accL;
  __syncthreads();
  if (threadIdx.x == 0) {
    float s = 0.f;
#pragma unroll
    for (int w = 0; w < NW; ++w) s += wsum[w];
    partial[blockIdx.x] = s;
  }
}

__global__ void copula_reduce(const float* __restrict__ partial, int nPart,
                              float* __restrict__ out) {
  __shared__ float sh[256];
  float s = 0.f;
  for (int i = threadIdx.x; i < nPart; i += 256) s += partial[i];
  sh[threadIdx.x] = s;
  __syncthreads();
  for (int k = 128; k > 0; k >>= 1) {
    if ((int)threadIdx.x < k) sh[threadIdx.x] += sh[threadIdx.x + k];
    __syncthreads();
  }
  if (threadIdx.x == 0) out[0] = sh[0];
}

extern "C" void kernel_launch(void* const* d_in, const int* in_sizes, int n_in,
                              void* d_out, int out_size, void* d_ws, size_t ws_size,
                              hipStream_t stream) {
  const float* x   = (const float*)d_in[0];
  const float* t   = (const float*)d_in[1];
  const float* c   = (const float*)d_in[2];
  const float* Wt1 = (const float*)d_in[3];
  const float* bt1 = (const float*)d_in[4];
  const float* Wt2 = (const float*)d_in[5];
  const float* bt2 = (const float*)d_in[6];
  const float* Wc1 = (const float*)d_in[7];
  const float* bc1 = (const float*)d_in[8];
  const float* Wc2 = (const float*)d_in[9];
  const float* bc2 = (const float*)d_in[10];
  const float* shape_t = (const float*)d_in[11];
  const float* scale_t = (const float*)d_in[12];
  const float* shape_c = (const float*)d_in[13];
  const float* scale_c = (const float*)d_in[14];

  float* ws      = (float*)d_ws;
  float* wEff    = ws;        // 130 floats used
  float* partial = ws + 256;  // GRID floats

  const int N = in_sizes[1];  // t has N elements

  copula_prep<<<1, 64, 0, stream>>>(Wt1, bt1, Wt2, bt2, Wc1, bc1, Wc2, bc2, wEff);
  copula_main<<<GRID, BLK, 0, stream>>>(x, t, c, wEff,
                                        shape_t, scale_t, shape_c, scale_c,
                                        partial, N);
  copula_reduce<<<1, 256, 0, stream>>>(partial, GRID, (float*)d_out);
}